// ProgramLineEncoderModel_90615220011534
// MI455X (gfx1250) — compile-verified
//
#include <hip/hip_runtime.h>
#include <hip/hip_bf16.h>

typedef float v2f __attribute__((ext_vector_type(2)));
typedef float v8f __attribute__((ext_vector_type(8)));

#define XS_STRIDE 68    // 64 + 4 pad -> conflict-free LDS A-fragment reads
#define HS_STRIDE 1028  // 1024 + 4 pad
#define D_MODEL 512
#define BATCH 128

__device__ __forceinline__ v8f wmma_f32(v2f a, v2f b, v8f c) {
  // D = A(16x4, f32) * B(4x16, f32) + C(16x16, f32)  -> v_wmma_f32_16x16x4_f32
  return __builtin_amdgcn_wmma_f32_16x16x4_f32(
      /*neg_a=*/false, a, /*neg_b=*/false, b,
      /*c_mod=*/(short)0, c, /*reuse_a=*/false, /*reuse_b=*/false);
}

__global__ __launch_bounds__(256) void fused_line_encoder_kernel(
    const int*   __restrict__ cons,      // (S,B,4) flattened, g*4 + j
    const int*   __restrict__ pad_mask,  // (B,S)
    const float* __restrict__ obj_e,     // (N,B,512)
    const float* __restrict__ type_emb,  // (8,512)
    const float* __restrict__ dir_emb,   // (8,512)
    const float* __restrict__ W1,        // (2048,1024)
    const float* __restrict__ b1,        // (1024)
    const float* __restrict__ W2,        // (1024,512)
    const float* __restrict__ b2,        // (512)
    float*       __restrict__ out,       // (S,B,512)
    int S)
{
  extern __shared__ unsigned char smem[];
  float* xs = (float*)smem;                                        // 32*68*4   = 8704 B
  float* hs = (float*)(smem + 32 * XS_STRIDE * 4);                 // 32*1028*4 = 131584 B
  const float** rp = (const float**)(smem + 32 * XS_STRIDE * 4
                                          + 32 * HS_STRIDE * 4);   // 32*4 ptrs = 1024 B

  const int tid  = threadIdx.x;
  const int lane = tid & 31;
  const int wave = tid >> 5;
  const int lo   = lane & 15;
  const int hi   = lane >> 4;     // 0 or 1
  const int kb   = hi << 1;       // K sub-offset 0 or 2 (ISA A/B f32 layout)
  const int row0 = blockIdx.x * 32;

  // ---- per-row gather pointers; nullptr encodes "this 512-segment is zero" ----
  if (tid < 32) {
    int g = row0 + tid;
    int b = g & (BATCH - 1);
    int s = g >> 7;
    int t  = cons[g * 4 + 0];
    int qi = cons[g * 4 + 1];
    int ri = cons[g * 4 + 2];
    int di = cons[g * 4 + 3];
    bool pad    = pad_mask[b * S + s] != 0;
    bool se     = (t == 4) | (t == 5);
    bool orient = (t == 2) | (t == 3);
    rp[tid * 4 + 0] = pad               ? nullptr : (type_emb + (size_t)t * D_MODEL);
    rp[tid * 4 + 1] = (pad | se)        ? nullptr : (obj_e + ((size_t)qi * BATCH + b) * D_MODEL);
    rp[tid * 4 + 2] = (pad | se)        ? nullptr : (obj_e + ((size_t)ri * BATCH + b) * D_MODEL);
    rp[tid * 4 + 3] = (pad | se | orient) ? nullptr : (dir_emb + (size_t)di * D_MODEL);
  }
  __syncthreads();

  // ================= GEMM1: h(32x1024) = relu(x(32x2048) @ W1 + b1) =================
  v8f acc[2][8];
  #pragma unroll
  for (int mt = 0; mt < 2; ++mt)
    #pragma unroll
    for (int nt = 0; nt < 8; ++nt)
      acc[mt][nt] = (v8f){};

  const int n0 = wave * 128;   // this wave's h-column slice

  for (int seg = 0; seg < 4; ++seg) {
    for (int k0 = 0; k0 < D_MODEL; k0 += 64) {
      __syncthreads();
      // stage x chunk (32 rows x 64 cols) into LDS; null segment pointer -> zeros
      #pragma unroll
      for (int i = 0; i < 2; ++i) {
        int idx = tid + i * 256;          // 0..511 float4 slots
        int r   = idx >> 4;
        int c4  = (idx & 15) << 2;
        const float* p = rp[r * 4 + seg];
        float4 v = {0.f, 0.f, 0.f, 0.f};
        if (p) v = *(const float4*)(p + k0 + c4);
        *(float4*)&xs[r * XS_STRIDE + c4] = v;
      }
      __syncthreads();

      const int krow = seg * D_MODEL + k0;
      if (krow + 64 < 2048)
        __builtin_prefetch(&W1[(size_t)(krow + 64) * 1024 + n0 + lo], 0, 1);

      #pragma unroll 4
      for (int kk = 0; kk < 64; kk += 4) {
        v2f a0 = *(const v2f*)&xs[(lo)      * XS_STRIDE + kk + kb];
        v2f a1 = *(const v2f*)&xs[(lo + 16) * XS_STRIDE + kk + kb];
        const float* wrow = W1 + (size_t)(krow + kk + kb) * 1024;
        #pragma unroll
        for (int nt = 0; nt < 8; ++nt) {
          int col = n0 + nt * 16 + lo;
          v2f bf;
          bf.x = wrow[col];
          bf.y = wrow[1024 + col];
          acc[0][nt] = wmma_f32(a0, bf, acc[0][nt]);
          acc[1][nt] = wmma_f32(a1, bf, acc[1][nt]);
        }
      }
    }
  }

  // bias + relu, deposit h into LDS (C/D layout: lane hi-half holds M=v+8)
  #pragma unroll
  for (int nt = 0; nt < 8; ++nt) {
    int col = n0 + nt * 16 + lo;
    float bb = b1[col];
    #pragma unroll
    for (int mt = 0; mt < 2; ++mt)
      #pragma unroll
      for (int v = 0; v < 8; ++v) {
        int mrow = mt * 16 + v + hi * 8;
        float hv = acc[mt][nt][v] + bb;
        hs[mrow * HS_STRIDE + col] = hv > 0.f ? hv : 0.f;
      }
  }
  __syncthreads();

  // ================= GEMM2: out(32x512) = h(32x1024) @ W2 + b2 =================
  v8f acc2[2][4];
  #pragma unroll
  for (int mt = 0; mt < 2; ++mt)
    #pragma unroll
    for (int nt = 0; nt < 4; ++nt)
      acc2[mt][nt] = (v8f){};

  const int n0b = wave * 64;   // this wave's out-column slice

  __builtin_prefetch(&W2[n0b + lo], 0, 1);
  #pragma unroll 4
  for (int kk = 0; kk < 1024; kk += 4) {
    v2f a0 = *(const v2f*)&hs[(lo)      * HS_STRIDE + kk + kb];
    v2f a1 = *(const v2f*)&hs[(lo + 16) * HS_STRIDE + kk + kb];
    const float* wrow = W2 + (size_t)(kk + kb) * 512;
    #pragma unroll
    for (int nt = 0; nt < 4; ++nt) {
      int col = n0b + nt * 16 + lo;
      v2f bf;
      bf.x = wrow[col];
      bf.y = wrow[512 + col];
      acc2[0][nt] = wmma_f32(a0, bf, acc2[0][nt]);
      acc2[1][nt] = wmma_f32(a1, bf, acc2[1][nt]);
    }
  }

  #pragma unroll
  for (int nt = 0; nt < 4; ++nt) {
    int col = n0b + nt * 16 + lo;
    float bb = b2[col];
    #pragma unroll
    for (int mt = 0; mt < 2; ++mt)
      #pragma unroll
      for (int v = 0; v < 8; ++v) {
        int mrow = mt * 16 + v + hi * 8;
        out[(size_t)(row0 + mrow) * 512 + col] = acc2[mt][nt][v] + bb;
      }
  }
}

extern "C" void kernel_launch(void* const* d_in, const int* in_sizes, int n_in,
                              void* d_out, int out_size, void* d_ws, size_t ws_size,
                              hipStream_t stream) {
  const int*   cons     = (const int*)d_in[0];   // constraints (S,B,4)
  const int*   pad_mask = (const int*)d_in[1];   // mask (B,S)
  const float* obj_e    = (const float*)d_in[2]; // (N,B,512)
  const float* type_emb = (const float*)d_in[3]; // (8,512)
  const float* dir_emb  = (const float*)d_in[4]; // (8,512)
  const float* W1       = (const float*)d_in[5]; // (2048,1024)
  const float* b1       = (const float*)d_in[6]; // (1024)
  const float* W2       = (const float*)d_in[7]; // (1024,512)
  const float* b2       = (const float*)d_in[8]; // (512)
  float*       out      = (float*)d_out;         // (S,B,512)

  const int rows = in_sizes[0] / 4;              // S*B = 65536
  const int S    = rows / BATCH;                 // 512
  const int grid = rows / 32;                    // 2048 blocks of 32 rows

  const size_t smem = 32 * XS_STRIDE * 4   // x staging
                    + 32 * HS_STRIDE * 4   // h tile
                    + 32 * 4 * sizeof(void*); // row segment pointers

  fused_line_encoder_kernel<<<grid, 256, smem, stream>>>(
      cons, pad_mask, obj_e, type_emb, dir_emb, W1, b1, W2, b2, out, S);
}